// HyperNetDynamicLinear_29832842838310
// MI455X (gfx1250) — compile-verified
//
#include <hip/hip_runtime.h>
#include <hip/hip_bf16.h>

#define MAX_IN   1024
#define MAX_OUT  4096
#define LOW_RANK 64
#define N_ARCH   16
#define HYP_HID  128
#define M_TOTAL  (4 * 2048)   // 8192 rows of x

#define BM 128
#define BN 128
#define BK 32
#define LDS_STRIDE 36         // 32 + 4 pad: keeps float4 16B alignment, conflict-free frag reads

typedef __attribute__((ext_vector_type(2))) float v2f;
typedef __attribute__((ext_vector_type(8))) float v8f;

// ---------------- Step 1: h = relu(hyp_w1 @ e + b1), [128] ----------------
__global__ void hyp_hidden_kernel(const float* __restrict__ w1,
                                  const float* __restrict__ b1,
                                  const float* __restrict__ e,
                                  float* __restrict__ h) {
    int i = threadIdx.x;
    if (i < HYP_HID) {
        float acc = b1[i];
#pragma unroll
        for (int j = 0; j < N_ARCH; ++j)
            acc += w1[i * N_ARCH + j] * e[j];
        h[i] = acc > 0.0f ? acc : 0.0f;
    }
}

// ------------- Step 2: left = hyp_w2 @ h + b2, [MAX_OUT*LOW_RANK] ----------
__global__ void hyp_left_kernel(const float* __restrict__ w2,
                                const float* __restrict__ b2,
                                const float* __restrict__ h,
                                float* __restrict__ left) {
    __shared__ float hs[HYP_HID];
    int t = threadIdx.x;
    if (t < HYP_HID) hs[t] = h[t];
    __syncthreads();
    int o = blockIdx.x * blockDim.x + t;   // one output per thread, 262144 total
    const float* row = w2 + (size_t)o * HYP_HID;
    float acc = b2[o];
#pragma unroll 8
    for (int j = 0; j < HYP_HID; ++j)
        acc += row[j] * hs[j];
    left[o] = acc;
}

// ------- Step 3: weight = base_weight + left @ lr_factor, [4096,1024] ------
__global__ void build_weight_kernel(const float* __restrict__ base_w,
                                    const float* __restrict__ left,
                                    const float* __restrict__ lr,
                                    float* __restrict__ weight) {
    int idx = blockIdx.x * blockDim.x + threadIdx.x;   // 4,194,304 elements
    int o = idx >> 10;          // /MAX_IN
    int i = idx & (MAX_IN - 1);
    float acc = base_w[idx];
    const float* lrow = left + o * LOW_RANK;
#pragma unroll 8
    for (int r = 0; r < LOW_RANK; ++r)
        acc += lrow[r] * lr[r * MAX_IN + i];
    weight[idx] = acc;
}

// ------------- Step 4: y = x @ weight^T + bias via fp32 WMMA --------------
__global__ __launch_bounds__(256)
void gemm_wmma_f32_kernel(const float* __restrict__ x,
                          const float* __restrict__ W,     // [MAX_OUT, MAX_IN]
                          const float* __restrict__ bias,  // [MAX_OUT]
                          float* __restrict__ out) {       // [M_TOTAL, MAX_OUT]
    __shared__ float xs[BM * LDS_STRIDE];
    __shared__ float wsl[BN * LDS_STRIDE];

    const int t    = threadIdx.x;
    const int wave = t >> 5;          // 0..7
    const int lane = t & 31;
    const int half = lane >> 4;       // K-pair select (A/B) / M-half select (C)
    const int l16  = lane & 15;

    const int m_base = blockIdx.y * BM;
    const int n_base = blockIdx.x * BN;
    const int wm = (wave & 3) * 32;   // wave's M offset within block (4 waves)
    const int wn = (wave >> 2) * 64;  // wave's N offset within block (2 waves)

    v8f acc[2][4];
#pragma unroll
    for (int i = 0; i < 2; ++i)
#pragma unroll
        for (int j = 0; j < 4; ++j)
            acc[i][j] = (v8f){0.f, 0.f, 0.f, 0.f, 0.f, 0.f, 0.f, 0.f};

    const int ldrow = t >> 3;         // 0..31
    const int ldcol = (t & 7) * 4;    // 0,4,...,28

    for (int k0 = 0; k0 < MAX_IN; k0 += BK) {
        // Stage 128x32 x-tile and 128x32 w-tile into LDS (float4, coalesced)
#pragma unroll
        for (int r = 0; r < 4; ++r) {
            const int row = r * 32 + ldrow;
            const float4 xv = *(const float4*)(x + (size_t)(m_base + row) * MAX_IN + k0 + ldcol);
            *(float4*)(&xs[row * LDS_STRIDE + ldcol]) = xv;
            const float4 wv = *(const float4*)(W + (size_t)(n_base + row) * MAX_IN + k0 + ldcol);
            *(float4*)(&wsl[row * LDS_STRIDE + ldcol]) = wv;
        }
        __syncthreads();

#pragma unroll
        for (int kk = 0; kk < BK; kk += 4) {
            const int c = kk + half * 2;   // lanes 0-15: K0,K1 ; lanes 16-31: K2,K3
            v2f a[2], b[4];
#pragma unroll
            for (int i = 0; i < 2; ++i) {
                const int row = (wm + i * 16 + l16) * LDS_STRIDE + c;
                a[i].x = xs[row];
                a[i].y = xs[row + 1];
            }
#pragma unroll
            for (int j = 0; j < 4; ++j) {
                const int row = (wn + j * 16 + l16) * LDS_STRIDE + c;
                b[j].x = wsl[row];
                b[j].y = wsl[row + 1];
            }
#pragma unroll
            for (int i = 0; i < 2; ++i)
#pragma unroll
                for (int j = 0; j < 4; ++j)
                    acc[i][j] = __builtin_amdgcn_wmma_f32_16x16x4_f32(
                        false, a[i], false, b[j], (short)0, acc[i][j], false, false);
        }
        __syncthreads();
    }

    // Store: C/D layout = VGPR r -> row (r + 8*half), col l16
#pragma unroll
    for (int i = 0; i < 2; ++i) {
#pragma unroll
        for (int j = 0; j < 4; ++j) {
            const int gm0 = m_base + wm + i * 16 + half * 8;
            const int gn  = n_base + wn + j * 16 + l16;
            const float bv = bias[gn];
#pragma unroll
            for (int r = 0; r < 8; ++r)
                out[(size_t)(gm0 + r) * MAX_OUT + gn] = acc[i][j][r] + bv;
        }
    }
}

extern "C" void kernel_launch(void* const* d_in, const int* in_sizes, int n_in,
                              void* d_out, int out_size, void* d_ws, size_t ws_size,
                              hipStream_t stream) {
    const float* x      = (const float*)d_in[0];
    const float* arch   = (const float*)d_in[1];
    const float* base_w = (const float*)d_in[2];
    const float* base_b = (const float*)d_in[3];
    const float* lr     = (const float*)d_in[4];
    const float* w1     = (const float*)d_in[5];
    const float* b1     = (const float*)d_in[6];
    const float* w2     = (const float*)d_in[7];
    const float* b2     = (const float*)d_in[8];
    float* out = (float*)d_out;

    float* wsf    = (float*)d_ws;
    float* h      = wsf;                           // 128 floats (pad to 256)
    float* left   = wsf + 256;                     // 262,144 floats
    float* weight = left + MAX_OUT * LOW_RANK;     // 4,194,304 floats (~18 MB total)

    hyp_hidden_kernel<<<1, 128, 0, stream>>>(w1, b1, arch, h);
    hyp_left_kernel<<<(MAX_OUT * LOW_RANK) / 256, 256, 0, stream>>>(w2, b2, h, left);
    build_weight_kernel<<<(MAX_OUT * MAX_IN) / 256, 256, 0, stream>>>(base_w, left, lr, weight);

    dim3 grid(MAX_OUT / BN, M_TOTAL / BM);         // 32 x 64 blocks
    gemm_wmma_f32_kernel<<<grid, 256, 0, stream>>>(x, weight, base_b, out);

    (void)in_sizes; (void)n_in; (void)out_size; (void)ws_size;
}